// CTCLossNM_31619549233422
// MI455X (gfx1250) — compile-verified
//
#include <hip/hip_runtime.h>

// Problem constants (match reference)
#define NUMCLS 512
#define B_ 32
#define T_ 1600
#define C_ 513            // NUMCLS + 1, blank index == NUMCLS
#define S_ 200
#define L_ 401            // 2*S + 1
#define NEGF (-1.0e30f)
#define BLOCK_ 416        // 13 wave32s, >= L_
#define LOG2E_ 1.4426950408889634f
#define LN2_   0.6931471805599453f

// ---- CDNA5 async global->LDS path (guarded; fallback keeps it compiling) ----
#if defined(__gfx1250__) && __has_builtin(__builtin_amdgcn_global_load_async_to_lds_b32)
#define CTC_ASYNC 1
#else
#define CTC_ASYNC 0
#endif

#if CTC_ASYNC
typedef __attribute__((address_space(1))) int* gint_ptr;
typedef __attribute__((address_space(3))) int* lint_ptr;

__device__ __forceinline__ void async_f32_to_lds(const float* g, float* l) {
  __builtin_amdgcn_global_load_async_to_lds_b32(
      (gint_ptr)(void*)g, (lint_ptr)(void*)l, 0, 0);
}
#endif

__device__ __forceinline__ void wait_async0() {
#if CTC_ASYNC
#if __has_builtin(__builtin_amdgcn_s_wait_asynccnt)
  __builtin_amdgcn_s_wait_asynccnt(0);
#else
  asm volatile("s_wait_asynccnt 0" ::: "memory");
#endif
#endif
}

// Raw base-2 transcendentals (v_exp_f32 / v_log_f32), with safe fallbacks.
__device__ __forceinline__ float fast_exp2(float x) {
#if __has_builtin(__builtin_amdgcn_exp2f)
  return __builtin_amdgcn_exp2f(x);
#else
  return __expf(x * LN2_);
#endif
}
__device__ __forceinline__ float fast_log2(float x) {
#if __has_builtin(__builtin_amdgcn_logf)
  return __builtin_amdgcn_logf(x);
#else
  return __logf(x) * LOG2E_;
#endif
}

// Stage one 513-float log-prob row into an LDS buffer (async DMA on CDNA5).
__device__ __forceinline__ void stage_row(const float* __restrict__ src,
                                          float* dst, int tid) {
#if CTC_ASYNC
  for (int i = tid; i < C_; i += BLOCK_) async_f32_to_lds(src + i, dst + i);
#else
  for (int i = tid; i < C_; i += BLOCK_) dst[i] = src[i];
#endif
}

// One workgroup per batch element; thread l owns extended-label position l for
// the whole scan. alpha (in log2 domain) is double-buffered in LDS with a
// 2-element NEG pad in front so alpha[l-1]/alpha[l-2] reads need no branch.
// Row t+1 is async-prefetched into the other LDS row buffer during step t.
// The reference freezes alpha at t >= input_length, so the scan ends there.
__global__ __launch_bounds__(BLOCK_) void ctc_alpha_kernel(
    const float* __restrict__ log_probs,      // [B, T, C]
    const int*   __restrict__ targets,        // [B, S]
    const int*   __restrict__ input_length,   // [B]
    const int*   __restrict__ target_length,  // [B]
    float*       __restrict__ nll_out) {      // [B]
  __shared__ float row[2][C_];
  __shared__ float alpha[2][L_ + 2];

  const int b   = blockIdx.x;
  const int tid = threadIdx.x;
  const int inlen = input_length[b];   // uniform within block; >= 800
  const int tlen  = target_length[b];

  // Per-thread static state: extended label + skip-transition permission.
  int  ext_l = NUMCLS;   // even positions (and idle lanes) -> blank
  bool skip  = false;
  if (tid < L_ && (tid & 1)) {
    const int j = (tid - 1) >> 1;
    ext_l = targets[b * S_ + j];
    if (tid >= 3) skip = (ext_l != targets[b * S_ + j - 1]);
  }

  // NEG pads at indices 0,1 of both alpha buffers; never rewritten.
  if (tid < 2) { alpha[0][tid] = NEGF; alpha[1][tid] = NEGF; }

  const float* base = log_probs + (size_t)b * T_ * C_;

  // Stage row 0, init alpha0 (scaled into log2 domain).
  stage_row(base, row[0], tid);
  wait_async0();
  __syncthreads();

  if (tid < L_) alpha[0][tid + 2] = (tid < 2) ? row[0][ext_l] * LOG2E_ : NEGF;

  // Prefetch row 1 (inlen >= 800, always needed).
  stage_row(base + C_, row[1], tid);
  wait_async0();
  __syncthreads();

  int cur = 1;  // row buffer holding lp_t at loop top
  int p   = 0;  // alpha buffer holding alpha_{t-1} at loop top
#pragma unroll 2
  for (int t = 1; t < inlen; ++t) {
    // Kick off the next row's DMA; it overlaps the transcendental chain below.
    if (t + 1 < inlen) stage_row(base + (size_t)(t + 1) * C_, row[cur ^ 1], tid);

    if (tid < L_) {
      const float* ap = alpha[p];
      const float a0 = ap[tid + 2];
      const float a1 = ap[tid + 1];
      const float a2 = skip ? ap[tid] : NEGF;
      const float m  = fmaxf(fmaxf(a0, a1), a2);
      const float s  = fast_exp2(a0 - m) + fast_exp2(a1 - m) + fast_exp2(a2 - m);
      alpha[p ^ 1][tid + 2] = fmaf(row[cur][ext_l], LOG2E_, m + fast_log2(s));
    }

    wait_async0();     // prefetch issued at loop top has landed
    __syncthreads();   // alpha writes + everyone's async portion visible
    cur ^= 1;
    p   ^= 1;
  }

  if (tid == 0) {
    const float a1 = alpha[p][2 * tlen - 1 + 2];   // log2 domain
    const float a2 = alpha[p][2 * tlen + 2];
    const float m  = fmaxf(a1, a2);
    const float l2 = m + fast_log2(fast_exp2(a1 - m) + fast_exp2(a2 - m));
    nll_out[b] = -l2 * LN2_;                       // back to natural log
  }
}

// Mean over B=32 per-sample losses with a single wave32.
__global__ void ctc_mean_kernel(const float* __restrict__ nll,
                                float* __restrict__ out) {
  float v = nll[threadIdx.x];
  for (int off = 16; off > 0; off >>= 1) v += __shfl_down(v, off, 32);
  if (threadIdx.x == 0) out[0] = v * (1.0f / (float)B_);
}

extern "C" void kernel_launch(void* const* d_in, const int* in_sizes, int n_in,
                              void* d_out, int out_size, void* d_ws, size_t ws_size,
                              hipStream_t stream) {
  (void)in_sizes; (void)n_in; (void)out_size; (void)ws_size;
  const float* log_probs     = (const float*)d_in[0];
  const int*   targets       = (const int*)d_in[1];
  const int*   input_length  = (const int*)d_in[2];
  const int*   target_length = (const int*)d_in[3];
  float*       nll           = (float*)d_ws;   // B_ floats of scratch

  ctc_alpha_kernel<<<B_, BLOCK_, 0, stream>>>(log_probs, targets, input_length,
                                              target_length, nll);
  ctc_mean_kernel<<<1, 32, 0, stream>>>(nll, (float*)d_out);
}